// SVDpp_55929064128604
// MI455X (gfx1250) — compile-verified
//
#include <hip/hip_runtime.h>

typedef __attribute__((ext_vector_type(2))) float v2f;
typedef __attribute__((ext_vector_type(8))) float v8f;

#define GLOBAL_MEAN 3.5f
#define D 64

// ---------------------------------------------------------------------------
// Kernel 1: u_impl_fdb[d] = (sum_h yj_emb[Iu[h]][d]) / sqrt(HIST)   (d = 0..63)
// One block of 64 threads; 200 coalesced 256B row reads, trivially fast.
// ---------------------------------------------------------------------------
__global__ void svdpp_implfdb_kernel(const int* __restrict__ Iu,
                                     const float* __restrict__ yj_emb,
                                     float* __restrict__ fdb, int hist) {
    const int d = threadIdx.x;            // 0..63
    float s = 0.0f;
    for (int h = 0; h < hist; ++h) {
        const long long it = Iu[h];
        s += yj_emb[(size_t)it * D + d];
    }
    fdb[d] = s * (1.0f / sqrtf((float)hist));
}

// ---------------------------------------------------------------------------
// Kernel 2: per-row rating via WMMA f32 16x16x4 "ones-column" batched dot.
//   Wave handles 16 batch rows. A[m][k] = (u[m][k]+f[k]) * v[m][k],
//   B[:,0] = 1 else 0  ->  D[m][0] = dot(user_vec[m], item_vec[m]).
// A layout (32-bit 16x4): lane l<16 -> (M=l, K=4c+{0,1}); lane l+16 -> K=4c+{2,3}.
// D layout: D[m][0] in lane 0 (m=0..7) / lane 16 (m=8..15), VGPRs 0..7.
// ---------------------------------------------------------------------------
__global__ __launch_bounds__(256) void svdpp_rating_kernel(
    const int* __restrict__ user_idx, const int* __restrict__ item_idx,
    const float* __restrict__ user_emb, const float* __restrict__ item_emb,
    const float* __restrict__ user_bias, const float* __restrict__ item_bias,
    const float* __restrict__ fdb, float* __restrict__ out, int B) {

    __shared__ __align__(16) float sF[D];   // broadcast feedback vector
    __shared__ float sDot[128];             // 16 dots per wave, 8 waves

    const int tid = threadIdx.x;
    if (tid < D) sF[tid] = fdb[tid];
    __syncthreads();

    const int wave  = tid >> 5;
    const int lane  = tid & 31;
    const int laneM = lane & 15;            // row-within-tile (M)
    const int half  = lane >> 4;            // K sub-pair selector

    int row = blockIdx.x * 128 + wave * 16 + laneM;
    if (row >= B) row = B - 1;              // keep EXEC all-ones for WMMA

    const int uidx = user_idx[row];
    const int iidx = item_idx[row];

    const float* up = user_emb + (size_t)uidx * D + half * 2;
    const float* ip = item_emb + (size_t)iidx * D + half * 2;
    const float* fp = sF + half * 2;

    // B matrix: column 0 all ones -> lane with N==0 holds (1,1), rest (0,0).
    const float bval = (laneM == 0) ? 1.0f : 0.0f;
    v2f bsel; bsel.x = bval; bsel.y = bval;

    v8f acc = {};
#pragma unroll
    for (int c = 0; c < 16; ++c) {          // 16 chunks of K=4 -> K=64 total
        const float2 u = *(const float2*)(up + c * 4);
        const float2 v = *(const float2*)(ip + c * 4);
        const float2 f = *(const float2*)(fp + c * 4);
        v2f a;
        a.x = (u.x + f.x) * v.x;
        a.y = (u.y + f.y) * v.y;
        acc = __builtin_amdgcn_wmma_f32_16x16x4_f32(
            /*neg_a=*/false, a, /*neg_b=*/false, bsel,
            /*c_mod=*/(short)0, acc, /*reuse_a=*/false, /*reuse_b=*/false);
    }

    // Lanes 0 / 16 hold the 16 dots in acc[0..7]; stage via LDS.
    if (laneM == 0) {
        const int base = wave * 16 + half * 8;
#pragma unroll
        for (int j = 0; j < 8; ++j) sDot[base + j] = acc[j];
    }
    __syncthreads();

    if (half == 0) {
        const float dot = sDot[wave * 16 + laneM];
        out[row] = dot + GLOBAL_MEAN + user_bias[uidx] + item_bias[iidx];
    }
}

// ---------------------------------------------------------------------------
extern "C" void kernel_launch(void* const* d_in, const int* in_sizes, int n_in,
                              void* d_out, int out_size, void* d_ws, size_t ws_size,
                              hipStream_t stream) {
    const int*   user_idx  = (const int*)d_in[0];
    const int*   item_idx  = (const int*)d_in[1];
    const int*   Iu        = (const int*)d_in[2];
    const float* user_emb  = (const float*)d_in[3];
    const float* item_emb  = (const float*)d_in[4];
    const float* user_bias = (const float*)d_in[5];
    const float* item_bias = (const float*)d_in[6];
    const float* yj_emb    = (const float*)d_in[7];
    float*       out       = (float*)d_out;

    const int B    = in_sizes[0];
    const int hist = in_sizes[2];
    float* fdb = (float*)d_ws;   // 64 floats of scratch

    svdpp_implfdb_kernel<<<1, D, 0, stream>>>(Iu, yj_emb, fdb, hist);

    const int blocks = (B + 127) / 128;   // 128 rows per 256-thread block
    svdpp_rating_kernel<<<blocks, 256, 0, stream>>>(
        user_idx, item_idx, user_emb, item_emb,
        user_bias, item_bias, fdb, out, B);
}